// Draw_contour_34763465294603
// MI455X (gfx1250) — compile-verified
//
#include <hip/hip_runtime.h>

#define SIZE   256
#define NPIX   (SIZE*SIZE)
#define NIMG   4
#define KP     64
#define KSIGN  100000.0f
#define EPSC   1e-5f
#define INV2PI 0.15915494309189533577f

typedef float v2f __attribute__((ext_vector_type(2)));
typedef float v8f __attribute__((ext_vector_type(8)));

// ---------------------------------------------------------------------------
// Kernel 1: winding-number soft rasterization of the contour.
// One block per (image, row); one thread per pixel column. Contour staged in
// LDS; per-edge sqrt reused across consecutive edges (norm_n(k) == norm_d(k+1)).
// ---------------------------------------------------------------------------
__global__ __launch_bounds__(SIZE) void mask_kernel(const float* __restrict__ contour,
                                                    float* __restrict__ mask) {
  __shared__ float2 c[KP];
  const int b = blockIdx.x >> 8;    // image
  const int i = blockIdx.x & 255;   // row
  const int j = threadIdx.x;        // col
  if (threadIdx.x == 0) __builtin_prefetch(contour + b * KP * 2, 0, 0); // global_prefetch_b8
  if (threadIdx.x < KP) {
    const float* src = contour + b * KP * 2 + threadIdx.x * 2;
    c[threadIdx.x] = make_float2(src[0], src[1]);
  }
  __syncthreads();

  const float px = (float)i * (1.0f / SIZE);   // mesh comp 0 = ii/size
  const float py = (float)j * (1.0f / SIZE);   // mesh comp 1 = jj/size

  float d0x = c[0].x - px, d0y = c[0].y - py;
  float n0  = sqrtf(d0x * d0x + d0y * d0y);
  float sum = 0.0f;
#pragma unroll 4
  for (int k = 0; k < KP; ++k) {
    const int   k1  = (k + 1) & (KP - 1);
    const float d1x = c[k1].x - px, d1y = c[k1].y - py;
    const float n1  = sqrtf(d1x * d1x + d1y * d1y);
    const float cross = d0y * d1x - d0x * d1y;           // diff.y*nxt.x - diff.x*nxt.y
    const float dotv  = d0x * d1x + d0y * d1y;
    float v = dotv / (n0 * n1);
    v = fminf(fmaxf(v, -1.0f + EPSC), 1.0f - EPSC);
    sum += tanhf(KSIGN * cross) * acosf(v);
    d0x = d1x; d0y = d1y; n0 = n1;
  }
  float s = fabsf(sum) * INV2PI;
  s = fminf(fmaxf(s, 0.0f), 1.0f);
  mask[b * NPIX + i * SIZE + j] = s;
}

// ---------------------------------------------------------------------------
// Kernel 2: horizontal Sobel pass as banded matmul Out = In x H using
// V_WMMA_F32_16X16X4_F32. One wave per 16x16 output tile; both horizontal
// filters ([1,0,-1] for Gx, [2,4,2] for Gy) accumulated simultaneously.
// Band sparsity: the central k-block needs 4 K-steps; the left neighbor block
// contributes only its last column (K-step s=3, row kk=3 -> b.y of hi=1), the
// right neighbor only its first column (s=0, kk=0 -> b.x of hi=0). Zero
// padding falls out of band truncation at the matrix edge.
// A layout (ISA 7.12.2, 32-bit A 16x4): lane<16 -> {K0,K1}, lane>=16 -> {K2,K3}.
// B layout (4x16): b.x holds row K=2*hi, b.y row K=2*hi+1, N = lane&15.
// C/D: VGPR v -> M = v + 8*(lane>>4), N = lane&15.
// ---------------------------------------------------------------------------
__global__ __launch_bounds__(256) void hpass_wmma_kernel(const float* __restrict__ mask,
                                                         float* __restrict__ A1,
                                                         float* __restrict__ A2) {
  const int wave = (blockIdx.x * blockDim.x + threadIdx.x) >> 5;
  const int lane = threadIdx.x & 31;
  const int img  = wave >> 8;        // 256 tiles per image
  const int tl   = wave & 255;
  const int tr   = tl >> 4;          // tile row-block
  const int tc   = tl & 15;          // tile col-block
  const int lrow = lane & 15;
  const int hi   = lane >> 4;
  const int tc_u = __builtin_amdgcn_readfirstlane(tc);  // wave-uniform -> scalar branches

  const float* src  = mask + img * NPIX;
  const int    r    = tr * 16 + lrow;   // A-matrix row (M = lane&15)
  const int    jcol = tc * 16 + lrow;   // B/D column  (N = lane&15)
  const float* rowp = src + r * SIZE;

  v8f acc1 = {0.f,0.f,0.f,0.f,0.f,0.f,0.f,0.f};
  v8f acc2 = {0.f,0.f,0.f,0.f,0.f,0.f,0.f,0.f};

  // Central k-block (tc): 4 K-steps; B depends only on (s, lane).
#pragma unroll
  for (int s = 0; s < 4; ++s) {
    const int cA = tc * 16 + s * 4 + 2 * hi;
    v2f a;
    a.x = rowp[cA];
    a.y = rowp[cA + 1];
    const int d0 = s * 4 + 2 * hi - lrow;   // (input col) - (output col)
    const int d1 = d0 + 1;
    v2f b1, b2;
    b1.x = (d0 == -1) ? 1.0f : (d0 == 1) ? -1.0f : 0.0f;   // [1,0,-1]
    b1.y = (d1 == -1) ? 1.0f : (d1 == 1) ? -1.0f : 0.0f;
    b2.x = (d0 == 0) ? 4.0f : ((d0 == -1) | (d0 == 1)) ? 2.0f : 0.0f;  // [2,4,2]
    b2.y = (d1 == 0) ? 4.0f : ((d1 == -1) | (d1 == 1)) ? 2.0f : 0.0f;
    acc1 = __builtin_amdgcn_wmma_f32_16x16x4_f32(false, a, false, b1,
                                                 (short)0, acc1, false, false);
    acc2 = __builtin_amdgcn_wmma_f32_16x16x4_f32(false, a, false, b2,
                                                 (short)0, acc2, false, false);
  }

  // Left neighbor block: only column p = 16*tc - 1 (K-step s=3, kk=3).
  if (tc_u > 0) {
    const int cA = tc * 16 - 4 + 2 * hi;   // Kbase = 16*(tc-1) + 12
    v2f a;
    a.x = rowp[cA];
    a.y = rowp[cA + 1];
    const float w = ((hi == 1) & (lrow == 0)) ? 1.0f : 0.0f; // H[16tc-1, 16tc] hit
    v2f b1 = {0.0f, w};          // Hx coeff +1
    v2f b2 = {0.0f, 2.0f * w};   // Hh coeff  2
    acc1 = __builtin_amdgcn_wmma_f32_16x16x4_f32(false, a, false, b1,
                                                 (short)0, acc1, false, false);
    acc2 = __builtin_amdgcn_wmma_f32_16x16x4_f32(false, a, false, b2,
                                                 (short)0, acc2, false, false);
  }

  // Right neighbor block: only column p = 16*tc + 16 (K-step s=0, kk=0).
  if (tc_u < 15) {
    const int cA = tc * 16 + 16 + 2 * hi;  // Kbase = 16*(tc+1)
    v2f a;
    a.x = rowp[cA];
    a.y = rowp[cA + 1];
    const float w = ((hi == 0) & (lrow == 15)) ? 1.0f : 0.0f; // H[16tc+16, 16tc+15] hit
    v2f b1 = {-w, 0.0f};         // Hx coeff -1
    v2f b2 = {2.0f * w, 0.0f};   // Hh coeff  2
    acc1 = __builtin_amdgcn_wmma_f32_16x16x4_f32(false, a, false, b1,
                                                 (short)0, acc1, false, false);
    acc2 = __builtin_amdgcn_wmma_f32_16x16x4_f32(false, a, false, b2,
                                                 (short)0, acc2, false, false);
  }

  const int orow0 = tr * 16 + hi * 8;
#pragma unroll
  for (int v = 0; v < 8; ++v) {
    const int idx = img * NPIX + (orow0 + v) * SIZE + jcol;
    A1[idx] = acc1[v];
    A2[idx] = acc2[v];
  }
}

// ---------------------------------------------------------------------------
// Kernel 3: vertical Sobel taps + magnitude + border zeroing.
// gx = 2*A1[i-1] + 4*A1[i] + 2*A1[i+1]; gy = A2[i-1] - A2[i+1] (zero pad).
// ---------------------------------------------------------------------------
__global__ __launch_bounds__(SIZE) void vpass_kernel(const float* __restrict__ A1,
                                                     const float* __restrict__ A2,
                                                     float* __restrict__ mag) {
  const int b = blockIdx.x >> 8;
  const int i = blockIdx.x & 255;
  const int j = threadIdx.x;
  const int idx = b * NPIX + i * SIZE + j;

  const float a1c = A1[idx];
  float a1m = 0.f, a1p = 0.f, a2m = 0.f, a2p = 0.f;
  if (i > 0)        { a1m = A1[idx - SIZE]; a2m = A2[idx - SIZE]; }
  if (i < SIZE - 1) { a1p = A1[idx + SIZE]; a2p = A2[idx + SIZE]; }

  const float gx = 2.0f * a1m + 4.0f * a1c + 2.0f * a1p;
  const float gy = a2m - a2p;
  float m = sqrtf(fmaxf(gx * gx + gy * gy, 1e-24f));
  if (i == 0 || i == SIZE - 1 || j == 0 || j == SIZE - 1) m = 0.0f;
  mag[idx] = m;
}

// ---------------------------------------------------------------------------
// Kernel 4: fused 3x3 maxpool (clipped window == -inf padding) + per-image
// min/max reduction + normalization, one 1024-thread block per image.
// ---------------------------------------------------------------------------
__global__ __launch_bounds__(1024) void pool_norm_kernel(const float* __restrict__ mag,
                                                         float* __restrict__ out) {
  const int b    = blockIdx.x;
  const int base = b * NPIX;
  __shared__ float smn[1024];
  __shared__ float smx[1024];

  for (int t = threadIdx.x; t < NPIX; t += 1024) {
    const int i = t >> 8, j = t & 255;
    const int i0 = (i > 0) ? i - 1 : 0, i1 = (i < SIZE - 1) ? i + 1 : SIZE - 1;
    const int j0 = (j > 0) ? j - 1 : 0, j1 = (j < SIZE - 1) ? j + 1 : SIZE - 1;
    float m = -__builtin_inff();
    for (int ii = i0; ii <= i1; ++ii)
      for (int jj = j0; jj <= j1; ++jj)
        m = fmaxf(m, mag[base + ii * SIZE + jj]);
    out[base + t] = m;
  }
  __threadfence_block();
  __syncthreads();

  float mn = __builtin_inff(), mx = -__builtin_inff();
  for (int t = threadIdx.x; t < NPIX; t += 1024) {
    const float v = out[base + t];
    mn = fminf(mn, v);
    mx = fmaxf(mx, v);
  }
  smn[threadIdx.x] = mn;
  smx[threadIdx.x] = mx;
  __syncthreads();
  for (int s = 512; s > 0; s >>= 1) {
    if (threadIdx.x < (unsigned)s) {
      smn[threadIdx.x] = fminf(smn[threadIdx.x], smn[threadIdx.x + s]);
      smx[threadIdx.x] = fmaxf(smx[threadIdx.x], smx[threadIdx.x + s]);
    }
    __syncthreads();
  }
  const float mnv   = smn[0];
  const float scale = 1.0f / (smx[0] - mnv + 1e-9f);
  for (int t = threadIdx.x; t < NPIX; t += 1024)
    out[base + t] = (out[base + t] - mnv) * scale;
}

// ---------------------------------------------------------------------------
extern "C" void kernel_launch(void* const* d_in, const int* in_sizes, int n_in,
                              void* d_out, int out_size, void* d_ws, size_t ws_size,
                              hipStream_t stream) {
  (void)in_sizes; (void)n_in; (void)out_size; (void)ws_size;
  const float* contour = (const float*)d_in[0];   // (4,1,64,2) f32
  float* out = (float*)d_out;                     // (4,1,256,256) f32
  float* wsf = (float*)d_ws;                      // needs 3 MB scratch

  float* mask = wsf;                   // [0, 1MB)
  float* A1   = wsf + 1 * NIMG * NPIX; // [1MB, 2MB)
  float* A2   = wsf + 2 * NIMG * NPIX; // [2MB, 3MB)
  float* mag  = wsf;                   // alias mask (dead after hpass)

  mask_kernel      <<<NIMG * SIZE,      SIZE, 0, stream>>>(contour, mask);
  hpass_wmma_kernel<<<(NIMG * 256) / 8, 256,  0, stream>>>(mask, A1, A2);
  vpass_kernel     <<<NIMG * SIZE,      SIZE, 0, stream>>>(A1, A2, mag);
  pool_norm_kernel <<<NIMG,             1024, 0, stream>>>(mag, out);
}